// GINE_19301583028827
// MI455X (gfx1250) — compile-verified
//
#include <hip/hip_runtime.h>
#include <hip/hip_bf16.h>
#include <math.h>

// ---------------- WMMA f32 16x16x4 helpers ----------------
typedef float v2f __attribute__((ext_vector_type(2)));
typedef float v8f __attribute__((ext_vector_type(8)));
typedef unsigned int v4u __attribute__((ext_vector_type(4)));
typedef int v4i __attribute__((ext_vector_type(4)));
typedef int v8i __attribute__((ext_vector_type(8)));

__device__ __forceinline__ v8f wmma4(v2f a, v2f b, v8f c) {
  // 8 args: (neg_a, A, neg_b, B, c_mod, C, reuse_a, reuse_b)
  return __builtin_amdgcn_wmma_f32_16x16x4_f32(false, a, false, b, (short)0, c,
                                               false, false);
}

__device__ __forceinline__ float elu_f(float x) {
  return x > 0.f ? x : (expf(x) - 1.f);
}

// ---------------- Tensor Data Mover: 2D f32 tile -> LDS ----------------
// D# per CDNA5 ISA ch.8: group0 = {flags, lds_addr, global_addr, type},
// group1 = {data_size, tensor dims, tile dims, dim0 stride}, groups2/3 = 0.
__device__ __forceinline__ void tdm_load_2d_f32(void* lds_dst,
                                                const void* gsrc,
                                                unsigned rows, unsigned cols,
                                                unsigned row_stride_elems) {
  unsigned long long ga = (unsigned long long)(uintptr_t)gsrc;
  unsigned lds_off = (unsigned)(uintptr_t)lds_dst;  // generic low 32b = LDS addr
  v4u g0;
  g0.x = 1u;                                        // count=1 (valid user D#)
  g0.y = lds_off;                                   // lds_addr (bytes)
  g0.z = (unsigned)(ga & 0xFFFFFFFFu);              // global_addr[31:0]
  g0.w = (unsigned)((ga >> 32) & 0x01FFFFFFu) | 0x80000000u;  // ga[56:32]|type=2
  v8i g1;
  g1[0] = (int)(2u << 16);                          // data_size=2 (4 bytes)
  g1[1] = (int)((cols & 0xFFFFu) << 16);            // tensor_dim0[15:0]
  g1[2] = (int)(((cols >> 16) & 0xFFFFu) | ((rows & 0xFFFFu) << 16));
  g1[3] = (int)(((rows >> 16) & 0xFFFFu) | ((cols & 0xFFFFu) << 16)); // tile_dim0
  g1[4] = (int)(rows & 0xFFFFu);                    // tile_dim1 (tile_dim2=0)
  g1[5] = (int)row_stride_elems;                    // tensor_dim0_stride[31:0]
  g1[6] = 0;
  g1[7] = 0;
  v4i z4 = {0, 0, 0, 0};
#if defined(__clang_major__) && (__clang_major__ >= 23)
  v8i z8 = {0, 0, 0, 0, 0, 0, 0, 0};
  __builtin_amdgcn_tensor_load_to_lds(g0, g1, z4, z4, z8, 0);
#else
  __builtin_amdgcn_tensor_load_to_lds(g0, g1, z4, z4, 0);
#endif
}

// A-matrix loaders -----------------------------------------
struct AGlobal {           // A[m][k] = p[m*ld + k]
  const float* p; int ld;
  __device__ __forceinline__ float operator()(int m, int k) const {
    return p[m * ld + k];
  }
};
struct ASum {              // A[m][k] = p0[m*ld+k] + p1[m*ld+k]
  const float* p0; const float* p1; int ld;
  __device__ __forceinline__ float operator()(int m, int k) const {
    return p0[m * ld + k] + p1[m * ld + k];
  }
};
struct ALds {              // A tile in LDS, row stride ld
  const float* p; int ld;
  __device__ __forceinline__ float operator()(int m, int k) const {
    return p[m * ld + k];
  }
};
// B-matrix loaders -----------------------------------------
struct BDense {            // B[k][c] = w[k*ld + c]
  const float* w; int ld;
  __device__ __forceinline__ float operator()(int k, int c) const {
    return w[k * ld + c];
  }
};
struct BPad {              // zero-padded columns beyond ncol
  const float* w; int ld; int ncol;
  __device__ __forceinline__ float operator()(int k, int c) const {
    return c < ncol ? w[k * ld + c] : 0.f;
  }
};

// acc[NT] covers a 16 x (NT*16) output tile; K multiple of 4.
template <int K, int NT, typename AF, typename BF>
__device__ __forceinline__ void wmma_gemm(const AF& A, const BF& B, v8f* acc,
                                          int lane) {
  const int m  = lane & 15;
  const int kh = (lane >> 4) << 1;  // 0 or 2
#pragma unroll 4
  for (int kk = 0; kk < K; kk += 4) {
    v2f a;
    a.x = A(m, kk + kh + 0);
    a.y = A(m, kk + kh + 1);
#pragma unroll
    for (int n = 0; n < NT; ++n) {
      v2f b;
      b.x = B(kk + kh + 0, n * 16 + m);
      b.y = B(kk + kh + 1, n * 16 + m);
      acc[n] = wmma4(a, b, acc[n]);
    }
  }
}

// ---------------- kernels ----------------

__global__ void zerof_kernel(float* __restrict__ p, size_t n) {
  size_t i = (size_t)blockIdx.x * blockDim.x + threadIdx.x;
  size_t stride = (size_t)gridDim.x * blockDim.x;
  for (; i < n; i += stride) p[i] = 0.f;
}

// conv1 messages: m = relu(x[src] + ea@We1 + be1), scatter-add to agg1[dst]
__global__ void edge_msg1_kernel(const int* __restrict__ ei,
                                 const float* __restrict__ ea,
                                 const float* __restrict__ x,
                                 const float* __restrict__ We1,
                                 const float* __restrict__ be1,
                                 float* __restrict__ agg1, int nedges) {
  __shared__ float sW[8 * 32];
  __shared__ float sb[32];
  for (int i = threadIdx.x; i < 8 * 32; i += blockDim.x) sW[i] = We1[i];
  for (int i = threadIdx.x; i < 32; i += blockDim.x) sb[i] = be1[i];
  __syncthreads();
  int e = blockIdx.x * blockDim.x + threadIdx.x;
  if (e >= nedges) return;
  int s = ei[e];
  int d = ei[nedges + e];
  float eav[8];
#pragma unroll
  for (int k = 0; k < 8; ++k) eav[k] = ea[e * 8 + k];
  const float* xs = x + (size_t)s * 32;
  float* ag = agg1 + (size_t)d * 32;
#pragma unroll 8
  for (int j = 0; j < 32; ++j) {
    float m = sb[j];
#pragma unroll
    for (int k = 0; k < 8; ++k) m = fmaf(eav[k], sW[k * 32 + j], m);
    m += xs[j];
    m = fmaxf(m, 0.f);
    atomicAdd(&ag[j], m);
  }
}

// conv2 messages: m = relu(h1[src] + ea@We2 + be2), scatter-add to agg2[dst]
// We2 (8x128) staged to LDS with the Tensor Data Mover.
__global__ void edge_msg2_kernel(const int* __restrict__ ei,
                                 const float* __restrict__ ea,
                                 const float* __restrict__ h1,
                                 const float* __restrict__ We2,
                                 const float* __restrict__ be2,
                                 float* __restrict__ agg2, int nedges) {
  __shared__ float sW[8 * 128];
  __shared__ float sb[128];
  if (threadIdx.x < 32) {                 // wave 0 issues the DMA once
    tdm_load_2d_f32(sW, We2, 8, 128, 128);
    __builtin_amdgcn_s_wait_tensorcnt(0);
  }
  for (int i = threadIdx.x; i < 128; i += blockDim.x) sb[i] = be2[i];
  __syncthreads();
  int e = blockIdx.x * blockDim.x + threadIdx.x;
  if (e >= nedges) return;
  int s = ei[e];
  int d = ei[nedges + e];
  float eav[8];
#pragma unroll
  for (int k = 0; k < 8; ++k) eav[k] = ea[e * 8 + k];
  const float* hs = h1 + (size_t)s * 128;
  float* ag = agg2 + (size_t)d * 128;
#pragma unroll 4
  for (int j = 0; j < 128; ++j) {
    float m = sb[j];
#pragma unroll
    for (int k = 0; k < 8; ++k) m = fmaf(eav[k], sW[k * 128 + j], m);
    m += hs[j];
    m = fmaxf(m, 0.f);
    atomicAdd(&ag[j], m);
  }
}

#define LD_T 132  // padded LDS row stride (floats) for the t tile
#define MLP_WAVES 4

// conv1 node MLP: h1 = elu( elu((x+agg1)@W1a + b1a) @ W1b + b1b )
// W1b (128x128, 64KB) staged to LDS via TDM, shared by all 4 waves.
__global__ __launch_bounds__(128) void node_mlp1_kernel(
    const float* __restrict__ x, const float* __restrict__ agg1,
    const float* __restrict__ W1a, const float* __restrict__ b1a,
    const float* __restrict__ W1b, const float* __restrict__ b1b,
    float* __restrict__ h1, int nnodes) {
  __shared__ float tbuf[MLP_WAVES][16 * LD_T];
  __shared__ float swb[128 * 128];
  if (threadIdx.x < 32) {
    tdm_load_2d_f32(swb, W1b, 128, 128, 128);
    __builtin_amdgcn_s_wait_tensorcnt(0);
  }
  __syncthreads();

  const int wave = threadIdx.x >> 5;
  const int lane = threadIdx.x & 31;
  const int tile = blockIdx.x * MLP_WAVES + wave;
  const int row0 = tile * 16;
  if (row0 < nnodes) {
    const int colm = lane & 15;
    const int moff = (lane >> 4) << 3;  // 0 or 8

    // GEMM1: (x + agg1)[16x32] @ W1a[32x128]
    v8f acc[8] = {};
    ASum a1{x + (size_t)row0 * 32, agg1 + (size_t)row0 * 32, 32};
    BDense bw1{W1a, 128};
    wmma_gemm<32, 8>(a1, bw1, acc, lane);

    float* t = tbuf[wave];
#pragma unroll
    for (int n = 0; n < 8; ++n) {
      int col = n * 16 + colm;
      float bias = b1a[col];
#pragma unroll
      for (int p = 0; p < 8; ++p) {
        float v = acc[n][p] + bias;
        t[(p + moff) * LD_T + col] = elu_f(v);
      }
    }

    // GEMM2: t[16x128] @ W1b[128x128] (weights from LDS)
    v8f acc2[8] = {};
    ALds a2{t, LD_T};
    BDense bw2{swb, 128};
    wmma_gemm<128, 8>(a2, bw2, acc2, lane);

#pragma unroll
    for (int n = 0; n < 8; ++n) {
      int col = n * 16 + colm;
      float bias = b1b[col];
#pragma unroll
      for (int p = 0; p < 8; ++p) {
        float v = acc2[n][p] + bias;
        h1[(size_t)(row0 + p + moff) * 128 + col] = elu_f(v);  // outer elu
      }
    }
  }
}

// conv2 node MLP fused with global-add-pool:
// h2 = elu( elu((h1+agg2)@W2a + b2a) @ W2b + b2b ); pool[batch[n]] += h2[n]
__global__ __launch_bounds__(128) void node_mlp2_kernel(
    const float* __restrict__ h1, const float* __restrict__ agg2,
    const float* __restrict__ W2a, const float* __restrict__ b2a,
    const float* __restrict__ W2b, const float* __restrict__ b2b,
    const int* __restrict__ batch, float* __restrict__ pool, int nnodes) {
  __shared__ float tbuf[MLP_WAVES][16 * LD_T];
  __shared__ float swb[128 * 128];
  if (threadIdx.x < 32) {
    tdm_load_2d_f32(swb, W2b, 128, 128, 128);
    __builtin_amdgcn_s_wait_tensorcnt(0);
  }
  __syncthreads();

  const int wave = threadIdx.x >> 5;
  const int lane = threadIdx.x & 31;
  const int tile = blockIdx.x * MLP_WAVES + wave;
  const int row0 = tile * 16;
  if (row0 < nnodes) {
    const int colm = lane & 15;
    const int moff = (lane >> 4) << 3;

    v8f acc[8] = {};
    ASum a1{h1 + (size_t)row0 * 128, agg2 + (size_t)row0 * 128, 128};
    BDense bw1{W2a, 128};
    wmma_gemm<128, 8>(a1, bw1, acc, lane);

    float* t = tbuf[wave];
#pragma unroll
    for (int n = 0; n < 8; ++n) {
      int col = n * 16 + colm;
      float bias = b2a[col];
#pragma unroll
      for (int p = 0; p < 8; ++p) {
        float v = acc[n][p] + bias;
        t[(p + moff) * LD_T + col] = elu_f(v);
      }
    }

    v8f acc2[8] = {};
    ALds a2{t, LD_T};
    BDense bw2{swb, 128};
    wmma_gemm<128, 8>(a2, bw2, acc2, lane);

    int gi[8];
#pragma unroll
    for (int p = 0; p < 8; ++p) gi[p] = batch[row0 + p + moff];

#pragma unroll
    for (int n = 0; n < 8; ++n) {
      int col = n * 16 + colm;
      float bias = b2b[col];
#pragma unroll
      for (int p = 0; p < 8; ++p) {
        float v = acc2[n][p] + bias;
        v = elu_f(v);  // outer elu fused
        atomicAdd(&pool[(size_t)gi[p] * 128 + col], v);
      }
    }
  }
}

// head: out = relu(pool@Wl1 + bl1) @ Wl2 + bl2   (T = 12 columns)
__global__ __launch_bounds__(128) void pool_mlp_kernel(
    const float* __restrict__ pool, const float* __restrict__ Wl1,
    const float* __restrict__ bl1, const float* __restrict__ Wl2,
    const float* __restrict__ bl2, float* __restrict__ out, int ngraphs) {
  __shared__ float tbuf[MLP_WAVES][16 * LD_T];
  __shared__ float swb[128 * 128];
  if (threadIdx.x < 32) {
    tdm_load_2d_f32(swb, Wl1, 128, 128, 128);
    __builtin_amdgcn_s_wait_tensorcnt(0);
  }
  __syncthreads();

  const int wave = threadIdx.x >> 5;
  const int lane = threadIdx.x & 31;
  const int tile = blockIdx.x * MLP_WAVES + wave;
  const int row0 = tile * 16;
  if (row0 < ngraphs) {
    const int colm = lane & 15;
    const int moff = (lane >> 4) << 3;

    v8f acc[8] = {};
    AGlobal a1{pool + (size_t)row0 * 128, 128};
    BDense bw1{swb, 128};
    wmma_gemm<128, 8>(a1, bw1, acc, lane);

    float* t = tbuf[wave];
#pragma unroll
    for (int n = 0; n < 8; ++n) {
      int col = n * 16 + colm;
      float bias = bl1[col];
#pragma unroll
      for (int p = 0; p < 8; ++p) {
        float v = acc[n][p] + bias;
        t[(p + moff) * LD_T + col] = fmaxf(v, 0.f);
      }
    }

    v8f acc2[1] = {};
    ALds a2{t, LD_T};
    BPad bw2{Wl2, 12, 12};
    wmma_gemm<128, 1>(a2, bw2, acc2, lane);

    if (colm < 12) {
      float bias = bl2[colm];
#pragma unroll
      for (int p = 0; p < 8; ++p) {
        out[(size_t)(row0 + p + moff) * 12 + colm] = acc2[0][p] + bias;
      }
    }
  }
}

// ---------------- launch ----------------
extern "C" void kernel_launch(void* const* d_in, const int* in_sizes, int n_in,
                              void* d_out, int out_size, void* d_ws,
                              size_t ws_size, hipStream_t stream) {
  const float* x     = (const float*)d_in[0];
  const int*   ei    = (const int*)d_in[1];
  const float* ea    = (const float*)d_in[2];
  const int*   batch = (const int*)d_in[3];
  const float* We1 = (const float*)d_in[4];
  const float* be1 = (const float*)d_in[5];
  const float* W1a = (const float*)d_in[6];
  const float* b1a = (const float*)d_in[7];
  const float* W1b = (const float*)d_in[8];
  const float* b1b = (const float*)d_in[9];
  const float* We2 = (const float*)d_in[10];
  const float* be2 = (const float*)d_in[11];
  const float* W2a = (const float*)d_in[12];
  const float* b2a = (const float*)d_in[13];
  const float* W2b = (const float*)d_in[14];
  const float* b2b = (const float*)d_in[15];
  const float* Wl1 = (const float*)d_in[16];
  const float* bl1 = (const float*)d_in[17];
  const float* Wl2 = (const float*)d_in[18];
  const float* bl2 = (const float*)d_in[19];

  const int N = in_sizes[0] / 32;
  const int E = in_sizes[1] / 2;
  const int G = out_size / 12;

  float* ws   = (float*)d_ws;
  float* agg1 = ws;                              // N*32
  float* agg2 = agg1 + (size_t)N * 32;           // N*128
  float* pool = agg2 + (size_t)N * 128;          // G*128
  float* h1   = pool + (size_t)G * 128;          // N*128

  // zero the accumulators (agg1, agg2, pool are contiguous)
  size_t nz = (size_t)N * 32 + (size_t)N * 128 + (size_t)G * 128;
  zerof_kernel<<<4096, 256, 0, stream>>>(agg1, nz);

  edge_msg1_kernel<<<(E + 255) / 256, 256, 0, stream>>>(ei, ea, x, We1, be1,
                                                        agg1, E);

  int ntiles = (N + 15) / 16;
  int nblk   = (ntiles + MLP_WAVES - 1) / MLP_WAVES;
  node_mlp1_kernel<<<nblk, 32 * MLP_WAVES, 0, stream>>>(x, agg1, W1a, b1a, W1b,
                                                        b1b, h1, N);

  edge_msg2_kernel<<<(E + 255) / 256, 256, 0, stream>>>(ei, ea, h1, We2, be2,
                                                        agg2, E);

  node_mlp2_kernel<<<nblk, 32 * MLP_WAVES, 0, stream>>>(
      h1, agg2, W2a, b2a, W2b, b2b, batch, pool, N);

  int gtiles = (G + 15) / 16;
  int gblk   = (gtiles + MLP_WAVES - 1) / MLP_WAVES;
  pool_mlp_kernel<<<gblk, 32 * MLP_WAVES, 0, stream>>>(pool, Wl1, bl1, Wl2, bl2,
                                                       (float*)d_out, G);
}